// HBatchNorm_15504831939341
// MI455X (gfx1250) — compile-verified
//
#include <hip/hip_runtime.h>
#include <cmath>

#define NB 64
#define NR 4096
#define ND 129
#define NDP 144
#define EPSF 1e-7f

// workspace layout (in floats)
#define S_OFF   0                        // NB*NDP  : per-batch raw sums
#define MU_OFF  (S_OFF + NB*NDP)         // NB*NDP  : centroid mu
#define WV_OFF  (MU_OFF + NB*NDP)        // NB*NDP  : mu + bias_m
#define BM_OFF  (WV_OFF + NB*NDP)        // NDP     : bias_m (batch independent)
#define PD_OFF  (BM_OFF + NDP)           // NB      : 1 - ldot(mu, bias_m)
#define SC_OFF  (PD_OFF + NB)            // NB      : sqrt(weight/(var+1e-6))
#define VP_OFF  (SC_OFF + NB)            // NB*256  : variance partials

typedef __attribute__((ext_vector_type(2))) float v2f;
typedef __attribute__((ext_vector_type(8))) float v8f;

__device__ __forceinline__ float wave_sum(float v) {
  v += __shfl_xor(v, 16, 32);
  v += __shfl_xor(v, 8, 32);
  v += __shfl_xor(v, 4, 32);
  v += __shfl_xor(v, 2, 32);
  v += __shfl_xor(v, 1, 32);
  return v;   // all lanes hold the total (fixed-order, deterministic)
}

// Minkowski partial dot: lane l owns dims {l, l+32, l+64, l+96} (+128 for lane 0).
// Only lane0/k0 (dim 0) carries the negative sign.
__device__ __forceinline__ float mpart(const float* a, const float* b, int lane) {
  float t0 = a[0] * b[0];
  float p = (lane == 0) ? -t0 : t0;
  p += a[1] * b[1];
  p += a[2] * b[2];
  p += a[3] * b[3];
  p += a[4] * b[4];
  return p;
}

__device__ __forceinline__ void load_vec(const float* base, int lane, float* v) {
  v[0] = base[lane];
  v[1] = base[lane + 32];
  v[2] = base[lane + 64];
  v[3] = base[lane + 96];
  v[4] = (lane == 0) ? base[128] : 0.0f;
}

__device__ __forceinline__ void load_vec_nt(const float* base, int lane, float* v) {
  v[0] = __builtin_nontemporal_load(base + lane);
  v[1] = __builtin_nontemporal_load(base + lane + 32);
  v[2] = __builtin_nontemporal_load(base + lane + 64);
  v[3] = __builtin_nontemporal_load(base + lane + 96);
  v[4] = (lane == 0) ? __builtin_nontemporal_load(base + 128) : 0.0f;
}

__device__ __forceinline__ void store_vec_nt(float* base, int lane, const float* v) {
  __builtin_nontemporal_store(v[0], base + lane);
  __builtin_nontemporal_store(v[1], base + lane + 32);
  __builtin_nontemporal_store(v[2], base + lane + 64);
  __builtin_nontemporal_store(v[3], base + lane + 96);
  if (lane == 0) __builtin_nontemporal_store(v[4], base + 128);
}

// ---------------------------------------------------------------------------
// Kernel A: per-batch column sum s[b][d] = sum_n x[b][n][d] via f32 WMMA.
// A = ones(16x4) so C[i][j] = sum_k B[k][j] (layout-robust in K); the lane->N
// column mapping of C follows the documented C/D layout (N = lane % 16).
// Two independent accumulators break the WMMA C->C RAW chain.
// grid = (64 batches, 9 d-groups of 16 cols), block = 256 (8 full waves).
// ---------------------------------------------------------------------------
__global__ __launch_bounds__(256) void hbn_sum_wmma(const float* __restrict__ x,
                                                    float* __restrict__ ws) {
  const int b    = blockIdx.x;
  const int g    = blockIdx.y;
  const int wid  = threadIdx.x >> 5;
  const int lane = threadIdx.x & 31;
  const int half = lane >> 4;
  const int j    = lane & 15;
  const int d    = g * 16 + j;
  const int dc   = (d < ND) ? d : 0;          // clamped (keeps address valid)
  const float dm = (d < ND) ? 1.0f : 0.0f;    // arithmetic column mask (EXEC stays full)

  const float* xb = x + (size_t)b * NR * ND;

  v8f acc0 = {};
  v8f acc1 = {};
  v2f a;  a.x = 1.0f;  a.y = 1.0f;            // ones matrix

  // wave covers row chunks c = wid + 8*it -> rows n = 4*wid + 32*it .. +3
  for (int it = 0; it < 128; it += 2) {
    const int n0 = (wid + (it << 3)) << 2;
    const float* r0 = xb + (size_t)n0 * ND + dc;
    const float* r1 = r0 + (size_t)32 * ND;   // next chunk for this wave
    const int rsel = half << 1;               // lane-half picks a row pair
    v2f b0, b1;
    b0.x = dm * r0[(size_t)rsel * ND];
    b0.y = dm * r0[(size_t)(rsel + 1) * ND];
    b1.x = dm * r1[(size_t)rsel * ND];
    b1.y = dm * r1[(size_t)(rsel + 1) * ND];
    acc0 = __builtin_amdgcn_wmma_f32_16x16x4_f32(false, a, false, b0,
                                                 (short)0, acc0, false, false);
    acc1 = __builtin_amdgcn_wmma_f32_16x16x4_f32(false, a, false, b1,
                                                 (short)0, acc1, false, false);
  }

  // acc[0]: lanes 0..15 hold colsum for column d (row M=0); reduce 8 waves in LDS
  __shared__ float red[8 * 16];
  if (lane < 16) red[wid * 16 + j] = acc0[0] + acc1[0];
  __syncthreads();
  if (threadIdx.x < 16) {
    float t = 0.0f;
#pragma unroll
    for (int w = 0; w < 8; ++w) t += red[w * 16 + threadIdx.x];
    const int dd = g * 16 + (int)threadIdx.x;
    if (dd < ND) ws[S_OFF + b * NDP + dd] = t;
  }
}

// ---------------------------------------------------------------------------
// Kernel B: per-batch mu, bias_m, mu+bias_m, transport denominator.
// grid = 64, block = 32 (one wave).
// ---------------------------------------------------------------------------
__global__ __launch_bounds__(32) void hbn_finalize_mu(const float* __restrict__ bias,
                                                      float* __restrict__ ws) {
  const int b = blockIdx.x;
  const int l = threadIdx.x;

  float sv[5];
  load_vec(ws + S_OFF + b * NDP, l, sv);

  // mu = s * rsqrt(max(-ldot(s,s), EPS))
  const float ld_ss = wave_sum(mpart(sv, sv, l));
  const float inv   = rsqrtf(fmaxf(-ld_ss, EPSF));
  float mv[5];
#pragma unroll
  for (int k = 0; k < 5; ++k) mv[k] = sv[k] * inv;

  // bias_m = expmap(origin, [0, bias]); dim d>=1 maps to bias[d-1]
  float bv[5];
#pragma unroll
  for (int k = 0; k < 4; ++k) {
    const int d = l + 32 * k;
    bv[k] = (d == 0) ? 0.0f : bias[d - 1];
  }
  bv[4] = (l == 0) ? bias[ND - 2] : 0.0f;     // dim 128 -> bias[127]
  const float nb2 = wave_sum(mpart(bv, bv, l));   // time comp is 0, sign moot
  const float nb  = sqrtf(fmaxf(nb2, EPSF));
  const float shn = sinhf(nb) / nb;
  const float chn = coshf(nb);
  float bmv[5];
#pragma unroll
  for (int k = 0; k < 4; ++k) {
    const int d = l + 32 * k;
    bmv[k] = (d == 0) ? chn : shn * bv[k];
  }
  bmv[4] = (l == 0) ? shn * bv[4] : 0.0f;

  const float ld_mb = wave_sum(mpart(mv, bmv, l));

  float wvv[5];
#pragma unroll
  for (int k = 0; k < 5; ++k) wvv[k] = mv[k] + bmv[k];

  // stores (padded stride NDP; padding is never read back)
  float* mu_o = ws + MU_OFF + b * NDP;
  float* wv_o = ws + WV_OFF + b * NDP;
  float* bm_o = ws + BM_OFF;                  // identical across blocks: deterministic
#pragma unroll
  for (int k = 0; k < 4; ++k) {
    mu_o[l + 32 * k] = mv[k];
    wv_o[l + 32 * k] = wvv[k];
    bm_o[l + 32 * k] = bmv[k];
  }
  if (l == 0) {
    mu_o[128] = mv[4];
    wv_o[128] = wvv[4];
    bm_o[128] = bmv[4];
    ws[PD_OFF + b] = 1.0f - ld_mb;
  }
}

// ---------------------------------------------------------------------------
// Kernel C: variance partials. grid = (64, 32), block = 256 (wave per row),
// each global wave gw handles rows n = gw + 256*it, it = 0..15.
// ---------------------------------------------------------------------------
__global__ __launch_bounds__(256) void hbn_var_partial(const float* __restrict__ x,
                                                       float* __restrict__ ws) {
  const int b    = blockIdx.x;
  const int wid  = threadIdx.x >> 5;
  const int lane = threadIdx.x & 31;
  const int gw   = blockIdx.y * 8 + wid;      // 0..255

  float mv[5];
  load_vec(ws + MU_OFF + b * NDP, lane, mv);

  float acc = 0.0f;
  for (int it = 0; it < 16; ++it) {
    const int n = gw + (it << 8);
    const float* r = x + ((size_t)b * NR + n) * ND;
    if (it + 1 < 16)
      __builtin_prefetch(r + (size_t)256 * ND + lane, 0, 0);
    float xv[5];
    load_vec(r, lane, xv);
    const float al = fmaxf(-wave_sum(mpart(mv, xv, lane)), 1.0f + EPSF);
    const float dd = acoshf(al);
    acc += dd * dd;
  }
  if (lane == 0) ws[VP_OFF + b * 256 + gw] = acc;
}

// ---------------------------------------------------------------------------
// Kernel D: reduce 256 variance partials per batch -> scale. grid=64, block=256.
// ---------------------------------------------------------------------------
__global__ __launch_bounds__(256) void hbn_var_finalize(const float* __restrict__ weight,
                                                        float* __restrict__ ws) {
  const int b = blockIdx.x;
  __shared__ float sm[256];
  sm[threadIdx.x] = ws[VP_OFF + b * 256 + threadIdx.x];
  __syncthreads();
#pragma unroll
  for (int s = 128; s > 0; s >>= 1) {
    if (threadIdx.x < s) sm[threadIdx.x] += sm[threadIdx.x + s];
    __syncthreads();
  }
  if (threadIdx.x == 0) {
    const float var = sm[0] * (1.0f / (float)NR);
    ws[SC_OFF + b] = sqrtf(weight[0] / (var + 1e-6f));
  }
}

// ---------------------------------------------------------------------------
// Kernel E: main transform. grid = (64, 32), block = 256; each wave handles
// 16 rows (n = gw + 256*it) with per-batch constants hoisted out of the loop.
// Non-temporal x loads (last use) + non-temporal output stores keep x in L2.
// ---------------------------------------------------------------------------
__global__ __launch_bounds__(256) void hbn_transform(const float* __restrict__ x,
                                                     float* __restrict__ out,
                                                     const float* __restrict__ ws) {
  const int b    = blockIdx.x;
  const int wid  = threadIdx.x >> 5;
  const int lane = threadIdx.x & 31;
  const int gw   = blockIdx.y * 8 + wid;      // 0..255

  float mv[5], bmv[5], wvv[5];
  load_vec(ws + MU_OFF + b * NDP, lane, mv);
  load_vec(ws + BM_OFF,           lane, bmv);
  load_vec(ws + WV_OFF + b * NDP, lane, wvv);
  const float pd = ws[PD_OFF + b];
  const float sc = ws[SC_OFF + b];

  for (int it = 0; it < 16; ++it) {
    const int n = gw + (it << 8);
    const float* r = x   + ((size_t)b * NR + n) * ND;
    float*       o = out + ((size_t)b * NR + n) * ND;
    if (it + 1 < 16)
      __builtin_prefetch(r + (size_t)256 * ND + lane, 0, 0);

    float xv[5];
    load_vec_nt(r, lane, xv);

    // logmap(mu, x)
    const float al = fmaxf(-wave_sum(mpart(mv, xv, lane)), 1.0f + EPSF);
    float uv[5];
#pragma unroll
    for (int k = 0; k < 5; ++k) uv[k] = xv[k] - al * mv[k];
    const float nu = sqrtf(fmaxf(wave_sum(mpart(uv, uv, lane)), EPSF));
    const float f  = acoshf(al) / nu;
    float vv[5];
#pragma unroll
    for (int k = 0; k < 5; ++k) vv[k] = f * uv[k];

    // transport to bias_m, rescale
    const float coef = wave_sum(mpart(bmv, vv, lane)) / pd;
    float vt[5];
#pragma unroll
    for (int k = 0; k < 5; ++k) vt[k] = sc * (vv[k] + coef * wvv[k]);

    // expmap(bias_m, vt)
    const float nn = sqrtf(fmaxf(wave_sum(mpart(vt, vt, lane)), EPSF));
    const float ch = coshf(nn);
    const float sh = sinhf(nn) / nn;
    float ov[5];
#pragma unroll
    for (int k = 0; k < 5; ++k) ov[k] = ch * bmv[k] + sh * vt[k];

    store_vec_nt(o, lane, ov);
  }
}

extern "C" void kernel_launch(void* const* d_in, const int* in_sizes, int n_in,
                              void* d_out, int out_size, void* d_ws, size_t ws_size,
                              hipStream_t stream) {
  const float* x      = (const float*)d_in[0];   // [64, 4096, 129]
  const float* bias   = (const float*)d_in[1];   // [128]
  const float* weight = (const float*)d_in[2];   // scalar
  float* out = (float*)d_out;
  float* ws  = (float*)d_ws;

  dim3 gA(NB, 9);
  hbn_sum_wmma<<<gA, 256, 0, stream>>>(x, ws);

  hbn_finalize_mu<<<NB, 32, 0, stream>>>(bias, ws);

  dim3 gC(NB, 32);
  hbn_var_partial<<<gC, 256, 0, stream>>>(x, ws);

  hbn_var_finalize<<<NB, 256, 0, stream>>>(weight, ws);

  dim3 gE(NB, 32);
  hbn_transform<<<gE, 256, 0, stream>>>(x, out, ws);
}